// PathwayEncoder_89979564852002
// MI455X (gfx1250) — compile-verified
//
#include <hip/hip_runtime.h>
#include <math.h>

typedef __attribute__((ext_vector_type(2))) float v2f;
typedef __attribute__((ext_vector_type(8))) float v8f;

#define NEG_SLOPE 0.2f

__device__ __forceinline__ float lrelu(float a) {
    return a > 0.0f ? a : NEG_SLOPE * a;
}

__device__ __forceinline__ void atomicMaxF(float* addr, float v) {
    // classic sign-split trick; works when initialized to -inf
    if (v >= 0.0f) atomicMax((int*)addr, __float_as_int(v));
    else           atomicMin((unsigned int*)addr, __float_as_uint(v));
}

// ---------------- init ----------------
// i ranges over N*16 float4's for out1; small arrays piggyback on low i.
__global__ void init_ws_kernel(float* __restrict__ out1,
                               float* __restrict__ m1, float* __restrict__ den1,
                               float* __restrict__ m2, float* __restrict__ den2,
                               float* __restrict__ pool, float* __restrict__ cnt,
                               int N, int G) {
    int total = N * 16;                    // N*64 floats as float4
    int stride = gridDim.x * blockDim.x;
    float4 z = make_float4(0.f, 0.f, 0.f, 0.f);
    for (int i = blockIdx.x * blockDim.x + threadIdx.x; i < total; i += stride) {
        ((float4*)out1)[i] = z;
        if (i < 2 * N) { m1[i] = -INFINITY; den1[i] = 0.0f; }
        if (i < N)     { m2[i] = -INFINITY; den2[i] = 0.0f; }
        if (i < G * 64) pool[i] = 0.0f;
        if (i < G)      cnt[i] = 0.0f;
    }
}

__global__ void zero4_kernel(float4* __restrict__ p, int total4) {
    int stride = gridDim.x * blockDim.x;
    float4 z = make_float4(0.f, 0.f, 0.f, 0.f);
    for (int i = blockIdx.x * blockDim.x + threadIdx.x; i < total4; i += stride)
        p[i] = z;
}

// Stage a block's 16x64 output tile (4 waves x one 16x16 WMMA tile) in LDS,
// then write back the contiguous 1024-float region with b128 stores.
// LDS stride 68: half-wave D writes hit disjoint bank sets; b128 reads 16B-aligned.
__device__ __forceinline__ void store_tile(float* __restrict__ dst,   // = h + row0*64
                                           const v8f& acc, float* Ds,
                                           int lane, int wave) {
    constexpr int LDD = 68;
    int col = wave * 16 + (lane & 15);
    int rb = (lane >> 4) ? 8 : 0;
#pragma unroll
    for (int i = 0; i < 8; ++i)
        Ds[(rb + i) * LDD + col] = acc[i];
    __syncthreads();
    int tid = wave * 32 + lane;  // 0..127
#pragma unroll
    for (int half = 0; half < 2; ++half) {
        int j = tid * 4 + half * 512;     // 0..1020, step 4
        int r = j >> 6, cc = j & 63;
        float4 v = *(const float4*)&Ds[r * LDD + cc];
        *(float4*)&dst[j] = v;
    }
}

// ---------------- layer-1 GEMM: h1 = x @ W1  (K = 4, one WMMA per tile) ----
// block = 128 threads = 4 waves; wave w computes the 16x16 tile at cols w*16.
// Assumes N % 16 == 0 (N = 100000 -> 6250 tiles).
__global__ void gemm1_wmma(const float* __restrict__ x, const float* __restrict__ W1,
                           float* __restrict__ h1, int N) {
    constexpr int LDD = 68;
    __shared__ float Ds[16 * LDD];
    int lane = threadIdx.x & 31;
    int wave = threadIdx.x >> 5;           // column tile 0..3
    int row0 = blockIdx.x * 16;

    int r  = row0 + (lane & 15);
    int kb = (lane >> 4) * 2;              // lanes 0-15: k=0,1 ; lanes 16-31: k=2,3
    float2 axy = *(const float2*)(x + r * 4 + kb);   // b64, 8B-aligned
    v2f a; a.x = axy.x; a.y = axy.y;

    int c = wave * 16 + (lane & 15);
    v2f b;
    b.x = W1[kb * 64 + c];
    b.y = W1[(kb + 1) * 64 + c];

    v8f acc = {};
    acc = __builtin_amdgcn_wmma_f32_16x16x4_f32(false, a, false, b, (short)0, acc,
                                                false, false);
    store_tile(h1 + (size_t)row0 * 64, acc, Ds, lane, wave);
}

// ---------------- layer-2 GEMM: h2 = ELU(agg1 + bias1) @ W2  (K=64) --------
// block = 128 threads = 4 waves; activation staged once in LDS per 16-row block.
__global__ void gemm2_wmma(const float* __restrict__ agg1, const float* __restrict__ bias1,
                           const float* __restrict__ W2, float* __restrict__ h2, int N) {
    constexpr int LDA = 68;                // pad to avoid LDS bank conflicts
    constexpr int LDD = 68;
    __shared__ float As[16 * LDA];
    __shared__ float Ds[16 * LDD];
    int tid = threadIdx.x;
    int row0 = blockIdx.x * 16;

    for (int i = tid; i < 16 * 64; i += 128) {
        int rl = i >> 6, cc = i & 63;
        float v = agg1[(size_t)(row0 + rl) * 64 + cc] + bias1[cc];
        As[rl * LDA + cc] = v > 0.0f ? v : (__expf(v) - 1.0f);  // ELU
    }
    __syncthreads();

    int lane = tid & 31;
    int wave = tid >> 5;
    int rloc = lane & 15;
    int khalf = (lane >> 4) * 2;
    int c = wave * 16 + rloc;

    v8f acc = {};
#pragma unroll
    for (int kc = 0; kc < 16; ++kc) {
        int kb = kc * 4 + khalf;
        float2 af = *(const float2*)&As[rloc * LDA + kb];   // ds b64, 8B-aligned
        v2f a; a.x = af.x; a.y = af.y;
        v2f b;
        b.x = W2[kb * 64 + c];
        b.y = W2[(kb + 1) * 64 + c];
        acc = __builtin_amdgcn_wmma_f32_16x16x4_f32(false, a, false, b, (short)0, acc,
                                                    false, false);
    }
    store_tile(h2 + (size_t)row0 * 64, acc, Ds, lane, wave);
}

// ---------------- per-node attention logits (one wave per node) ------------
// Layer 1 (2 heads): lanes 0-15 cover channels 0..31 (head 0), lanes 16-31
// cover 32..63 (head 1); att_src/att_dst are flat [2*32] matching h layout.
__global__ void attn1_kernel(const float* __restrict__ h, const float* __restrict__ att_src,
                             const float* __restrict__ att_dst,
                             float* __restrict__ asrc, float* __restrict__ adst, int N) {
    int lane = threadIdx.x & 31;
    int n = blockIdx.x * (blockDim.x >> 5) + (threadIdx.x >> 5);
    if (n >= N) return;
    int c0 = lane * 2;
    float2 hv = *(const float2*)(h + (size_t)n * 64 + c0);
    float2 as = *(const float2*)(att_src + c0);
    float2 ad = *(const float2*)(att_dst + c0);
    float s = hv.x * as.x + hv.y * as.y;
    float d = hv.x * ad.x + hv.y * ad.y;
#pragma unroll
    for (int off = 1; off < 16; off <<= 1) {   // reduce within 16-lane half
        s += __shfl_xor(s, off, 32);
        d += __shfl_xor(d, off, 32);
    }
    if (lane == 0)  { asrc[n * 2 + 0] = s; adst[n * 2 + 0] = d; }
    if (lane == 16) { asrc[n * 2 + 1] = s; adst[n * 2 + 1] = d; }
}

__global__ void attn2_kernel(const float* __restrict__ h, const float* __restrict__ att_src,
                             const float* __restrict__ att_dst,
                             float* __restrict__ asrc, float* __restrict__ adst, int N) {
    int lane = threadIdx.x & 31;
    int n = blockIdx.x * (blockDim.x >> 5) + (threadIdx.x >> 5);
    if (n >= N) return;
    int c0 = lane * 2;
    float2 hv = *(const float2*)(h + (size_t)n * 64 + c0);
    float2 as = *(const float2*)(att_src + c0);
    float2 ad = *(const float2*)(att_dst + c0);
    float s = hv.x * as.x + hv.y * as.y;
    float d = hv.x * ad.x + hv.y * ad.y;
#pragma unroll
    for (int off = 1; off < 32; off <<= 1) {   // full-wave reduction
        s += __shfl_xor(s, off, 32);
        d += __shfl_xor(d, off, 32);
    }
    if (lane == 0) { asrc[n] = s; adst[n] = d; }
}

// ---------------- edge softmax passes (thread per edge) --------------------
template <int H>
__global__ void edge_max_kernel(const long long* __restrict__ srcIdx,
                                const long long* __restrict__ dstIdx, int E, int N,
                                const float* __restrict__ asrc, const float* __restrict__ adst,
                                float* __restrict__ m) {
    int total = E + N;
    int stride = gridDim.x * blockDim.x;
    for (int e = blockIdx.x * blockDim.x + threadIdx.x; e < total; e += stride) {
        int s, d;
        if (e < E) { s = (int)srcIdx[e]; d = (int)dstIdx[e]; }
        else       { s = e - E; d = s; }
        if (H == 2) {
            float2 as = *(const float2*)(asrc + (size_t)s * 2);
            float2 ad = *(const float2*)(adst + (size_t)d * 2);
            atomicMaxF(&m[d * 2 + 0], lrelu(as.x + ad.x));
            atomicMaxF(&m[d * 2 + 1], lrelu(as.y + ad.y));
        } else {
            atomicMaxF(&m[d], lrelu(asrc[s] + adst[d]));
        }
    }
}

template <int H>
__global__ void edge_sum_kernel(const long long* __restrict__ srcIdx,
                                const long long* __restrict__ dstIdx, int E, int N,
                                const float* __restrict__ asrc, const float* __restrict__ adst,
                                const float* __restrict__ m, float* __restrict__ den) {
    int total = E + N;
    int stride = gridDim.x * blockDim.x;
    for (int e = blockIdx.x * blockDim.x + threadIdx.x; e < total; e += stride) {
        int s, d;
        if (e < E) { s = (int)srcIdx[e]; d = (int)dstIdx[e]; }
        else       { s = e - E; d = s; }
        if (H == 2) {
            float2 as = *(const float2*)(asrc + (size_t)s * 2);
            float2 ad = *(const float2*)(adst + (size_t)d * 2);
            float2 mm = *(const float2*)(m + (size_t)d * 2);
            atomicAdd(&den[d * 2 + 0], __expf(lrelu(as.x + ad.x) - mm.x));
            atomicAdd(&den[d * 2 + 1], __expf(lrelu(as.y + ad.y) - mm.y));
        } else {
            atomicAdd(&den[d], __expf(lrelu(asrc[s] + adst[d]) - m[d]));
        }
    }
}

// one wave per edge: lane handles channels {2*lane, 2*lane+1}
// (both channels of a lane lie in the same head: head = c/32, lane<16 <=> c<32)
template <int H>
__global__ void edge_agg_kernel(const long long* __restrict__ srcIdx,
                                const long long* __restrict__ dstIdx, int E, int N,
                                const float* __restrict__ asrc, const float* __restrict__ adst,
                                const float* __restrict__ m, const float* __restrict__ den,
                                const float* __restrict__ hfeat, float* __restrict__ out) {
    int lane = threadIdx.x & 31;
    int wavesPerBlock = blockDim.x >> 5;
    int wave = blockIdx.x * wavesPerBlock + (threadIdx.x >> 5);
    int nW = gridDim.x * wavesPerBlock;
    int total = E + N;
    int c0 = lane * 2;
    for (int e = wave; e < total; e += nW) {
        int s, d;
        if (e < E) { s = (int)srcIdx[e]; d = (int)dstIdx[e]; }
        else       { s = e - E; d = s; }
        float alpha;
        if (H == 2) {
            float2 as = *(const float2*)(asrc + (size_t)s * 2);
            float2 ad = *(const float2*)(adst + (size_t)d * 2);
            float2 mm = *(const float2*)(m + (size_t)d * 2);
            float2 dn = *(const float2*)(den + (size_t)d * 2);
            float alpha0 = __expf(lrelu(as.x + ad.x) - mm.x) / (dn.x + 1e-16f);
            float alpha1 = __expf(lrelu(as.y + ad.y) - mm.y) / (dn.y + 1e-16f);
            alpha = (lane < 16) ? alpha0 : alpha1;
        } else {
            alpha = __expf(lrelu(asrc[s] + adst[d]) - m[d]) / (den[d] + 1e-16f);
        }
        float2 hv = *(const float2*)(hfeat + (size_t)s * 64 + c0);   // b64 gather
        atomicAdd(&out[(size_t)d * 64 + c0],     hv.x * alpha);
        atomicAdd(&out[(size_t)d * 64 + c0 + 1], hv.y * alpha);
    }
}

// ---------------- pooling (thread per 4 channels) --------------------------
__global__ void pool_sum_kernel(const float* __restrict__ feat, const long long* __restrict__ batch,
                                float* __restrict__ pool, float* __restrict__ cnt, int N) {
    int total = N * 16;
    int stride = gridDim.x * blockDim.x;
    for (int t = blockIdx.x * blockDim.x + threadIdx.x; t < total; t += stride) {
        int n = t >> 4, q = (t & 15) * 4;
        int b = (int)batch[n];
        float4 v = *(const float4*)(feat + (size_t)n * 64 + q);
        atomicAdd(&pool[b * 64 + q + 0], v.x);
        atomicAdd(&pool[b * 64 + q + 1], v.y);
        atomicAdd(&pool[b * 64 + q + 2], v.z);
        atomicAdd(&pool[b * 64 + q + 3], v.w);
        if (q == 0) atomicAdd(&cnt[b], 1.0f);
    }
}

__global__ void finalize_kernel(const float* __restrict__ pool, const float* __restrict__ cnt,
                                const float* __restrict__ bias2, float* __restrict__ out, int G) {
    int t = blockIdx.x * blockDim.x + threadIdx.x;
    if (t >= G * 64) return;
    int g = t >> 6, c = t & 63;
    out[t] = pool[t] / fmaxf(cnt[g], 1.0f) + bias2[c];
}

// ---------------- launcher ------------------------------------------------
extern "C" void kernel_launch(void* const* d_in, const int* in_sizes, int n_in,
                              void* d_out, int out_size, void* d_ws, size_t ws_size,
                              hipStream_t stream) {
    const float*      x        = (const float*)d_in[0];
    const long long*  ei       = (const long long*)d_in[1];
    const long long*  batch    = (const long long*)d_in[2];
    const float*      W1       = (const float*)d_in[3];
    const float*      att_src1 = (const float*)d_in[4];
    const float*      att_dst1 = (const float*)d_in[5];
    const float*      bias1    = (const float*)d_in[6];
    const float*      W2       = (const float*)d_in[7];
    const float*      att_src2 = (const float*)d_in[8];
    const float*      att_dst2 = (const float*)d_in[9];
    const float*      bias2    = (const float*)d_in[10];
    float*            out      = (float*)d_out;

    int N = in_sizes[0] / 4;       // 100000 (divisible by 16)
    int E = in_sizes[1] / 2;       // 3200000
    int G = out_size / 64;         // 64

    float* ws = (float*)d_ws;
    size_t off = 0;
    float* buf0  = ws + off; off += (size_t)N * 64;   // h1, later h2
    float* buf1  = ws + off; off += (size_t)N * 64;   // agg1, later agg2
    float* asrc1 = ws + off; off += (size_t)2 * N;
    float* adst1 = ws + off; off += (size_t)2 * N;
    float* m1    = ws + off; off += (size_t)2 * N;
    float* den1  = ws + off; off += (size_t)2 * N;
    float* asrc2 = ws + off; off += (size_t)N;
    float* adst2 = ws + off; off += (size_t)N;
    float* m2    = ws + off; off += (size_t)N;
    float* den2  = ws + off; off += (size_t)N;
    float* pool  = ws + off; off += (size_t)G * 64;
    float* cnt   = ws + off; off += (size_t)G;

    const long long* srcIdx = ei;
    const long long* dstIdx = ei + E;

    int rowTiles   = N / 16;                          // N % 16 == 0
    int edgeBlocks = ((E + N) + 255) / 256;           // thread per edge
    int aggBlocks  = ((E + N) + 8 - 1) / 8;           // 8 waves per 256-thread block
    int waveBlocks = (N + 7) / 8;                     // 8 node-waves per block
    int featBlocks = 4096;                            // grid-stride over N*16 float4

    // layer 1
    init_ws_kernel<<<featBlocks, 256, 0, stream>>>(buf1, m1, den1, m2, den2, pool, cnt, N, G);
    gemm1_wmma<<<rowTiles, 128, 0, stream>>>(x, W1, buf0, N);
    attn1_kernel<<<waveBlocks, 256, 0, stream>>>(buf0, att_src1, att_dst1, asrc1, adst1, N);
    edge_max_kernel<2><<<edgeBlocks, 256, 0, stream>>>(srcIdx, dstIdx, E, N, asrc1, adst1, m1);
    edge_sum_kernel<2><<<edgeBlocks, 256, 0, stream>>>(srcIdx, dstIdx, E, N, asrc1, adst1, m1, den1);
    edge_agg_kernel<2><<<aggBlocks, 256, 0, stream>>>(srcIdx, dstIdx, E, N, asrc1, adst1, m1, den1,
                                                      buf0, buf1);
    // layer 2 (ELU + bias fused into GEMM A-load; buf0 reused for h2)
    gemm2_wmma<<<rowTiles, 128, 0, stream>>>(buf1, bias1, W2, buf0, N);
    attn2_kernel<<<waveBlocks, 256, 0, stream>>>(buf0, att_src2, att_dst2, asrc2, adst2, N);
    zero4_kernel<<<featBlocks, 256, 0, stream>>>((float4*)buf1, N * 16);
    edge_max_kernel<1><<<edgeBlocks, 256, 0, stream>>>(srcIdx, dstIdx, E, N, asrc2, adst2, m2);
    edge_sum_kernel<1><<<edgeBlocks, 256, 0, stream>>>(srcIdx, dstIdx, E, N, asrc2, adst2, m2, den2);
    edge_agg_kernel<1><<<aggBlocks, 256, 0, stream>>>(srcIdx, dstIdx, E, N, asrc2, adst2, m2, den2,
                                                      buf0, buf1);
    // pooling (bias2 folded into the mean)
    pool_sum_kernel<<<featBlocks, 256, 0, stream>>>(buf1, batch, pool, cnt, N);
    finalize_kernel<<<(G * 64 + 255) / 256, 256, 0, stream>>>(pool, cnt, bias2, out, G);
}